// MhsLayer_31336081391791
// MI455X (gfx1250) — compile-verified
//
#include <hip/hip_runtime.h>
#include <hip/hip_bf16.h>

typedef __attribute__((ext_vector_type(2))) float v2f;
typedef __attribute__((ext_vector_type(8))) float v8f;
typedef __attribute__((ext_vector_type(4))) float f4;
typedef __attribute__((ext_vector_type(4))) int   i4;

#define NB   8
#define NL   1024
#define NIN  256
#define NOUT 4
#define NEGV (-1e-12f)

// Workspace layout (float units):
//   Weff : [256*16]      @ 0       (cols 0..3 = Wu@Wuv_top, 4..7 = Wv@Wuv_bot, 8..15 = 0)
//   beff : [16]          @ 4096
//   Pt   : [8*8*1024]    @ 4112    (Pt[(b*8+r)*1024 + l]; r<4: pu, r>=4: pv)
#define WS_BEFF 4096
#define WS_PT   4112

// ---------------------------------------------------------------------------
// Kernel A: fold Wuv into Wu/Wv.  Weff[k][r] = sum_c W[k][c] * Wuv_part[c][r]
// 4096 Weff entries + 16 beff entries; trivial cost.
// ---------------------------------------------------------------------------
__global__ void __launch_bounds__(256)
fold_weights(const float* __restrict__ Wu, const float* __restrict__ bu,
             const float* __restrict__ Wv, const float* __restrict__ bv,
             const float* __restrict__ Wuv, float* __restrict__ ws) {
    int gid = blockIdx.x * 256 + threadIdx.x;
    float* Weff = ws;
    float* beff = ws + WS_BEFF;
    if (gid < 256 * 16) {
        int k = gid >> 4, r = gid & 15;
        float acc = 0.f;
        if (r < 8) {
            const float* W  = (r < 4) ? Wu : Wv;
            const float* uv = Wuv + ((r < 4) ? 0 : NIN * NOUT);
            int rr = r & 3;
            for (int c = 0; c < NIN; ++c)
                acc = fmaf(W[k * NIN + c], uv[c * NOUT + rr], acc);
        }
        Weff[k * 16 + r] = acc;
    } else if (gid < 256 * 16 + 16) {
        int r = gid - 256 * 16;
        float acc = 0.f;
        if (r < 8) {
            const float* bb = (r < 4) ? bu : bv;
            const float* uv = Wuv + ((r < 4) ? 0 : NIN * NOUT);
            int rr = r & 3;
            for (int c = 0; c < NIN; ++c)
                acc = fmaf(bb[c], uv[c * NOUT + rr], acc);
        }
        beff[r] = acc;
    }
}

// ---------------------------------------------------------------------------
// Kernel B: P = inputs @ Weff + beff via V_WMMA_F32_16X16X4_F32.
// One wave per 16-row tile of the flattened (B*L, 256) input.
// Result stored transposed: Pt[(b*8 + n)*1024 + l] for coalesced reads in C.
// ---------------------------------------------------------------------------
__global__ void __launch_bounds__(256)
proj_wmma(const float* __restrict__ x, float* __restrict__ ws) {
    const float* Weff = ws;
    const float* beff = ws + WS_BEFF;
    float*       Pt   = ws + WS_PT;

    int lane = threadIdx.x & 31;
    int wave = threadIdx.x >> 5;
    int tile = blockIdx.x * 8 + wave;          // 0..511, 16 rows each

    int n     = lane & 15;                      // A: M row / B,D: column N
    int khalf = (lane >> 4) << 1;               // 0 for lanes 0-15, 2 for 16-31

    const float* arow = x + (size_t)(tile * 16 + n) * NIN;

    v8f acc = {};
    for (int kk = 0; kk < NIN; kk += 4) {
        v2f a, b;
        a.x = arow[kk + khalf];
        a.y = arow[kk + khalf + 1];
        b.x = Weff[(kk + khalf) * 16 + n];
        b.y = Weff[(kk + khalf + 1) * 16 + n];
        acc = __builtin_amdgcn_wmma_f32_16x16x4_f32(
            /*neg_a=*/false, a, /*neg_b=*/false, b,
            /*c_mod=*/(short)0, acc, /*reuse_a=*/false, /*reuse_b=*/false);
    }

    // bias along columns (lane's column is n)
    float bn = beff[n];
#pragma unroll
    for (int g = 0; g < 8; ++g) acc[g] += bn;

    // D layout: lanes 0-15 hold rows g, lanes 16-31 hold rows 8+g (col = lane&15).
    // Each lane's 8 values are 8 consecutive l's -> two 128-bit stores, transposed.
    if (n < 8) {
        int gm0 = tile * 16 + ((lane >> 4) << 3);   // first global row of this lane
        int b_  = gm0 >> 10;                        // batch (tile never straddles b)
        int l0  = gm0 & 1023;
        float* dst = Pt + (size_t)(b_ * 8 + n) * NL + l0;
        f4 lo = { acc[0], acc[1], acc[2], acc[3] };
        f4 hi = { acc[4], acc[5], acc[6], acc[7] };
        *(f4*)(dst)     = lo;
        *(f4*)(dst + 4) = hi;
    }
}

// ---------------------------------------------------------------------------
// Kernel C: the bandwidth kernel.
// out[b,r,i,j] = mask[b,i] && mask[b,j] ? Pt[b,4+r,i] + Pt[b,r,j] : -1e-12
// One block per (b,r,i) row; 256 threads x float4 = 1024 j's (4 KB store/block).
// ---------------------------------------------------------------------------
__global__ void __launch_bounds__(256)
bcast_mask(const float* __restrict__ ws, const int* __restrict__ mask,
           float* __restrict__ out) {
    const float* Pt = ws + WS_PT;

    int blk = blockIdx.x;          // 0 .. 8*4*1024-1
    int i   = blk & 1023;
    int r   = (blk >> 10) & 3;
    int b   = blk >> 12;
    int j0  = threadIdx.x << 2;

    bool  mi = mask[b * NL + i] != 0;
    float pv = Pt[(size_t)(b * 8 + 4 + r) * NL + i];

    f4 pu = *(const f4*)(Pt + (size_t)(b * 8 + r) * NL + j0);
    i4 mj = *(const i4*)(mask + b * NL + j0);

    f4 o;
    o.x = (mi && mj.x) ? pv + pu.x : NEGV;
    o.y = (mi && mj.y) ? pv + pu.y : NEGV;
    o.z = (mi && mj.z) ? pv + pu.z : NEGV;
    o.w = (mi && mj.w) ? pv + pu.w : NEGV;

    *(f4*)(out + (size_t)((b * NOUT + r) * NL + i) * NL + j0) = o;
}

// ---------------------------------------------------------------------------
extern "C" void kernel_launch(void* const* d_in, const int* in_sizes, int n_in,
                              void* d_out, int out_size, void* d_ws, size_t ws_size,
                              hipStream_t stream) {
    const float* inputs = (const float*)d_in[0];   // (8,1024,256) f32
    const int*   mask   = (const int*)  d_in[1];   // (8,1024) i32
    const float* Wu     = (const float*)d_in[2];   // (256,256)
    const float* bu     = (const float*)d_in[3];   // (256,)
    const float* Wv     = (const float*)d_in[4];   // (256,256)
    const float* bv     = (const float*)d_in[5];   // (256,)
    const float* Wuv    = (const float*)d_in[6];   // (512,4)
    float*       out    = (float*)d_out;           // (8,4,1024,1024)
    float*       ws     = (float*)d_ws;            // >= (4112 + 65536) * 4 bytes

    // A: fold weights (4096 Weff + 16 beff entries)
    fold_weights<<<17, 256, 0, stream>>>(Wu, bu, Wv, bv, Wuv, ws);
    // B: 512 tiles, 8 waves/block -> 64 blocks
    proj_wmma<<<64, 256, 0, stream>>>(inputs, ws);
    // C: 8*4*1024 rows
    bcast_mask<<<NB * NOUT * NL, 256, 0, stream>>>(ws, mask, out);
}